// ObjectCondensationLoss_30236569764496
// MI455X (gfx1250) — compile-verified
//
#include <hip/hip_runtime.h>
#include <hip/hip_bf16.h>

#define KINST 256
#define BATCH 4

typedef __attribute__((ext_vector_type(2))) float v2f;
typedef __attribute__((ext_vector_type(4))) float v4f;
typedef __attribute__((ext_vector_type(8))) float v8f;

// Workspace layout (re-initialized by oc_init every launch; ~16 KB total).
struct Ws {
  int      first_cp[BATCH * KINST];   // first CP hit index per instance (N = none)
  float    seg_sum [BATCH * KINST];   // attraction segment sums
  unsigned seg_cnt [BATCH * KINST];   // attraction segment counts
  float    cp_norm [BATCH * KINST];   // ||cp_k||^2
  float    pos_sum[BATCH], neg_sum[BATCH], rep_sum[BATCH];
  unsigned cpv_cnt[BATCH], noncp_cnt[BATCH], valid_cnt[BATCH];
};

__device__ __forceinline__ float wredf(float v) {
#pragma unroll
  for (int o = 16; o > 0; o >>= 1) v += __shfl_xor(v, o, 32);
  return v;
}
__device__ __forceinline__ unsigned wredu(unsigned v) {
#pragma unroll
  for (int o = 16; o > 0; o >>= 1) v += __shfl_xor(v, o, 32);
  return v;
}
// Numerically stable softplus: log(1+exp(x))
__device__ __forceinline__ float softplusf(float x) {
  return fmaxf(x, 0.f) + log1pf(expf(-fabsf(x)));
}

__global__ void oc_init(Ws* __restrict__ ws, int N) {
  int t = blockIdx.x * blockDim.x + threadIdx.x;
  if (t < BATCH * KINST) {
    ws->first_cp[t] = N;
    ws->seg_sum[t]  = 0.f;
    ws->seg_cnt[t]  = 0u;
    ws->cp_norm[t]  = 0.f;
  }
  if (t < BATCH) {
    ws->pos_sum[t] = 0.f; ws->neg_sum[t] = 0.f; ws->rep_sum[t] = 0.f;
    ws->cpv_cnt[t] = 0u;  ws->noncp_cnt[t] = 0u; ws->valid_cnt[t] = 0u;
  }
}

// Pass 1: beta loss partial sums, counts, and first_cp (segment-min of index).
// One block covers 256 points of a single batch -> one atomic per quantity per block.
__global__ void oc_scan(const float* __restrict__ beta,
                        const int* __restrict__ sid,
                        const unsigned char* __restrict__ iscp,
                        Ws* __restrict__ ws, int N) {
  int bpb = N / 256;                       // blocks per batch
  int b   = blockIdx.x / bpb;
  int n   = (blockIdx.x % bpb) * 256 + threadIdx.x;
  size_t g = (size_t)b * N + n;

  int   s    = sid[g];
  bool  cp   = iscp[g] != 0;
  float bt   = beta[g];
  bool  valid = s >= 0;
  bool  cpv   = cp && valid;

  float pos = cpv   ? softplusf(-bt) : 0.f;   // BCE target 1
  float neg = (!cp) ? softplusf(bt)  : 0.f;   // BCE target 0

  if (cpv) {
    int sc = s < KINST ? s : KINST - 1;
    atomicMin(&ws->first_cp[b * KINST + sc], n);
  }

  float    pr = wredf(pos), nr = wredf(neg);
  unsigned cr = wredu(cpv ? 1u : 0u);
  unsigned orr = wredu(cp ? 0u : 1u);
  unsigned vr = wredu(valid ? 1u : 0u);

  __shared__ float    sp[8], sn[8];
  __shared__ unsigned scw[8], sow[8], svw[8];
  int wid = threadIdx.x >> 5, lane = threadIdx.x & 31;
  if (lane == 0) { sp[wid] = pr; sn[wid] = nr; scw[wid] = cr; sow[wid] = orr; svw[wid] = vr; }
  __syncthreads();
  if (threadIdx.x == 0) {
    float P = 0.f, Ng = 0.f; unsigned C = 0, O = 0, V = 0;
    for (int i = 0; i < 8; i++) { P += sp[i]; Ng += sn[i]; C += scw[i]; O += sow[i]; V += svw[i]; }
    atomicAdd(&ws->pos_sum[b], P);
    atomicAdd(&ws->neg_sum[b], Ng);
    atomicAdd(&ws->cpv_cnt[b], C);
    atomicAdd(&ws->noncp_cnt[b], O);
    atomicAdd(&ws->valid_cnt[b], V);
  }
}

// ||cp_k||^2 : one wave per (batch, instance).
__global__ void oc_cpnorm(const float* __restrict__ embed, Ws* __restrict__ ws,
                          int N, int D) {
  int w    = (blockIdx.x * blockDim.x + threadIdx.x) >> 5;
  int lane = threadIdx.x & 31;
  if (w >= BATCH * KINST) return;
  int b = w / KINST, k = w % KINST;
  int fc  = ws->first_cp[b * KINST + k];
  int fcc = fc < N ? fc : N - 1;
  const float* row = embed + ((size_t)b * N + fcc) * (size_t)D;
  float acc = 0.f;
  for (int d = lane * 4; d < D; d += 128) {
    v4f x = *(const v4f*)(row + d);
    acc += x.x * x.x + x.y * x.y + x.z * x.z + x.w * x.w;
  }
  acc = wredf(acc);
  if (lane == 0) ws->cp_norm[b * KINST + k] = acc;
}

// Attraction: one wave per point; streams 256 MB of embed (bandwidth-bound).
// Point row loaded nontemporal (touched once); CP row stays hot in L2.
__global__ void oc_attract(const float* __restrict__ embed,
                           const int* __restrict__ sid,
                           Ws* __restrict__ ws, int N, int D) {
  int w    = (blockIdx.x * blockDim.x + threadIdx.x) >> 5;  // global wave == point
  int lane = threadIdx.x & 31;
  if (w >= BATCH * N) return;
  int b = w / N, n = w % N;
  int s = sid[(size_t)b * N + n];
  if (s < 0) return;                                        // invalid -> no weight
  int sc = s < KINST ? s : KINST - 1;
  int fc = ws->first_cp[b * KINST + sc];
  if (fc >= N) return;                                      // instance has no CP
  const float* e = embed + ((size_t)b * N + n)  * (size_t)D;
  const float* c = embed + ((size_t)b * N + fc) * (size_t)D;
  float acc = 0.f;
  for (int d = lane * 4; d < D; d += 128) {
    v4f ev = __builtin_nontemporal_load((const v4f*)(e + d));
    v4f cv = *(const v4f*)(c + d);
    v4f t  = ev - cv;
    acc += t.x * t.x + t.y * t.y + t.z * t.z + t.w * t.w;
  }
  acc = wredf(acc);
  if (lane == 0) {
    atomicAdd(&ws->seg_sum[b * KINST + sc], acc);
    atomicAdd(&ws->seg_cnt[b * KINST + sc], 1u);
  }
}

// Repulsion: Gram matrix G = C C^T per batch via V_WMMA_F32_16X16X4_F32.
// One wave per 16x16 output tile; D=256 -> 64 chained WMMA ops (K=4 each).
// ISA 7.12.2 f32 A-frag 16x4: lanes 0-15 hold K={0,1}, lanes 16-31 hold K={2,3},
// M = lane&15 -> one aligned float2 load per fragment per K-step. B (4x16) uses
// the identical addressing with the j-side row (C rows on both sides of C C^T).
__global__ void oc_repulse(const float* __restrict__ embed, Ws* __restrict__ ws,
                           int N, int D) {
  const int TPB = (KINST / 16) * (KINST / 16);   // 256 tiles per batch
  int w    = (blockIdx.x * blockDim.x + threadIdx.x) >> 5;
  int lane = threadIdx.x & 31;
  if (w >= BATCH * TPB) return;
  int b  = w / TPB, t = w % TPB;
  int i0 = (t / (KINST / 16)) * 16;
  int j0 = (t % (KINST / 16)) * 16;
  int hi = lane >> 4;           // 0: K-pair {0,1}, 1: K-pair {2,3}
  int lm = lane & 15;

  int fa = ws->first_cp[b * KINST + i0 + lm]; fa = fa < N ? fa : N - 1;
  int fb = ws->first_cp[b * KINST + j0 + lm]; fb = fb < N ? fb : N - 1;
  const float* rowA = embed + ((size_t)b * N + fa) * (size_t)D;
  const float* rowB = embed + ((size_t)b * N + fb) * (size_t)D;

  v8f acc = {0.f, 0.f, 0.f, 0.f, 0.f, 0.f, 0.f, 0.f};
  for (int kk = 0; kk < D / 4; ++kk) {
    int base = kk * 4 + hi * 2;
    v2f a  = *(const v2f*)(rowA + base);
    v2f bb = *(const v2f*)(rowB + base);
#if defined(__gfx1250__)
    // 8 args: (neg_a, A, neg_b, B, c_mod, C, reuse_a, reuse_b)
    acc = __builtin_amdgcn_wmma_f32_16x16x4_f32(
        false, a, false, bb, (short)0, acc, false, false);
#else
    acc[0] += a[0] * bb[0] + a[1] * bb[1];   // host/non-gfx1250 parse-only path
#endif
  }

  // C/D layout: VGPR r -> M = i0 + r + 8*hi ; N = j0 + (lane&15)
  int   j   = j0 + lm;
  bool  hcj = ws->first_cp[b * KINST + j] < N;
  float nj  = ws->cp_norm[b * KINST + j];
  float ssum = 0.f;
#pragma unroll
  for (int r = 0; r < 8; ++r) {
    int   i   = i0 + r + 8 * hi;
    bool  hci = ws->first_cp[b * KINST + i] < N;
    float ni  = ws->cp_norm[b * KINST + i];
    float pd2 = ni + nj - 2.f * acc[r];
    if (hci && hcj) ssum += expf(-pd2);
  }
  ssum = wredf(ssum);
  if (lane == 0) atomicAdd(&ws->rep_sum[b], ssum);
}

// Final epilogue: exact reference arithmetic, single block of K threads.
__global__ void oc_finalize(Ws* __restrict__ ws, float* __restrict__ out, int N) {
  int tid = threadIdx.x;                 // == instance k
  int lane = tid & 31, wid = tid >> 5;
  __shared__ float    sf[8];
  __shared__ unsigned su[8];
  float total = 0.f; unsigned okc = 0u;
  for (int b = 0; b < BATCH; ++b) {
    float cnt  = (float)ws->seg_cnt[b * KINST + tid];
    float ssum = ws->seg_sum[b * KINST + tid];
    float a_k  = cnt > 0.f ? ssum / fmaxf(cnt, 1.f) : 0.f;
    unsigned hc = ws->first_cp[b * KINST + tid] < N ? 1u : 0u;
    float ar = wredf(a_k); unsigned hr = wredu(hc);
    if (lane == 0) { sf[wid] = ar; su[wid] = hr; }
    __syncthreads();
    if (tid == 0) {
      float att = 0.f; unsigned M = 0u;
      for (int i = 0; i < 8; i++) { att += sf[i]; M += su[i]; }
      unsigned cpv = ws->cpv_cnt[b], noncp = ws->noncp_cnt[b], val = ws->valid_cnt[b];
      float pos = ws->pos_sum[b] / (float)(cpv   > 1u ? cpv   : 1u);
      float neg = ws->neg_sum[b] / (float)(noncp > 1u ? noncp : 1u);
      float beta_loss = 1.0f * pos + 0.5f * neg;            // BETA_POS_W, BETA_NEG_W
      float mm  = (float)M * (float)M;
      float rep = ws->rep_sum[b] / fmaxf(mm, 1.f);
      rep = (M > 1u) ? rep * 1.0f : 0.f;                    // REPULSION_W
      float loss = beta_loss + att * 1.0f + rep;            // ATTRACTION_W
      bool ok = (val > 0u) && (cpv > 0u);
      total += ok ? loss : 0.f;
      okc   += ok ? 1u : 0u;
    }
    __syncthreads();
  }
  if (tid == 0) out[0] = okc > 0u ? total / (float)okc : 0.f;
}

extern "C" void kernel_launch(void* const* d_in, const int* in_sizes, int n_in,
                              void* d_out, int out_size, void* d_ws, size_t ws_size,
                              hipStream_t stream) {
  const float*         beta  = (const float*)d_in[0];
  const float*         embed = (const float*)d_in[1];
  const int*           sid   = (const int*)d_in[2];
  const unsigned char* iscp  = (const unsigned char*)d_in[3];
  float* out = (float*)d_out;
  Ws*    ws  = (Ws*)d_ws;

  const int BN = in_sizes[0];        // B*N (beta is (B,N,1))
  const int N  = BN / BATCH;         // 65536
  const int D  = in_sizes[1] / BN;   // 256

  oc_init    <<<(BATCH * KINST + 255) / 256, 256, 0, stream>>>(ws, N);
  oc_scan    <<<BN / 256,                    256, 0, stream>>>(beta, sid, iscp, ws, N);
  oc_cpnorm  <<<(BATCH * KINST) / 8,         256, 0, stream>>>(embed, ws, N, D);
  oc_attract <<<BN / 8,                      256, 0, stream>>>(embed, sid, ws, N, D);
  oc_repulse <<<(BATCH * (KINST / 16) * (KINST / 16)) / 8, 256, 0, stream>>>(embed, ws, N, D);
  oc_finalize<<<1, KINST, 0, stream>>>(ws, out, N);
}